// SOM_6408091206116
// MI455X (gfx1250) — compile-verified
//
#include <hip/hip_runtime.h>
#include <hip/hip_bf16.h>

typedef float v2f __attribute__((ext_vector_type(2)));
typedef float v8f __attribute__((ext_vector_type(8)));

#define D 512            // feature dim (fixed by reference)
#define TOPK 16

// Extract element r (0..7) of an 8-wide f32 accumulator with a constant select chain.
__device__ __forceinline__ float pick8(v8f c, int r) {
    float v = c[0];
    if (r == 1) v = c[1];
    if (r == 2) v = c[2];
    if (r == 3) v = c[3];
    if (r == 4) v = c[4];
    if (r == 5) v = c[5];
    if (r == 6) v = c[6];
    if (r == 7) v = c[7];
    return v;
}

// Phase 1: each wave computes, for a 16-row tile of the map,
//   partial_dist2[row] = ||m_row||^2 - 2 * dot(m_row, s)
// using V_WMMA_F32_16X16X4_F32:
//   cdot  = A x B_s  (B_s = sample replicated in all 16 columns -> every column holds dot_i)
//   cgram = A x A    (A's register image IS its own transpose's B image -> diagonal = row norms)
__global__ __launch_bounds__(256) void som_dist2_kernel(
    const float* __restrict__ samples,
    const float* __restrict__ map,
    float* __restrict__ out_d2,
    int ntiles)
{
    const int wave = threadIdx.x >> 5;
    const int lane = threadIdx.x & 31;
    const int tile = blockIdx.x * 8 + wave;
    if (tile >= ntiles) return;                 // wave-uniform: EXEC stays all-ones for WMMA

    const int row = lane & 15;                  // A: lanes 0-15 and 16-31 both map M=0..15
    const int hi  = (lane >> 4) & 1;            // lanes 16-31 hold K+2,K+3
    const float* arow = map + (size_t)(tile * 16 + row) * D + hi * 2;
    const float* brow = samples + hi * 2;       // B_s: N-independent (all columns = sample)

    v8f cdot  = {};
    v8f cgram = {};

    for (int k0 = 0; k0 < D; k0 += 32) {
        // stay ~1KB ahead of the A stream
        __builtin_prefetch(arow + k0 + 256, 0, 1);
        #pragma unroll
        for (int k = k0; k < k0 + 32; k += 4) {
            v2f a = *(const v2f*)(arow + k);    // global_load_b64 per lane
            v2f b = *(const v2f*)(brow + k);    // hits cache (2KB sample)
            cdot  = __builtin_amdgcn_wmma_f32_16x16x4_f32(
                        false, a, false, b, (short)0, cdot,  false, false);
            cgram = __builtin_amdgcn_wmma_f32_16x16x4_f32(
                        false, a, false, a, (short)0, cgram, false, false);
        }
    }

    // Diagonal C[i][i]: i in 0..7 -> (lane i, vgpr i); i in 8..15 -> (lane i+16, vgpr i-8)
    int myrow = -1;
    if (lane < 8)        myrow = lane;
    else if (lane >= 24) myrow = lane - 16;
    if (myrow >= 0) {
        int r   = myrow & 7;
        float g = pick8(cgram, r);              // ||m_row||^2
        float d = pick8(cdot,  r);              // dot(m_row, s)
        out_d2[tile * 16 + myrow] = g - 2.0f * d;
    }
}

// Phase 2: single block top-16 (smallest) over N dist^2-proxies.
__global__ __launch_bounds__(256) void som_topk_kernel(
    const float* __restrict__ samples,
    const float* __restrict__ d2,
    int N,
    float* __restrict__ out)   // [0..15]: int32 indices (reinterpreted), [16..31]: f32 dists
{
    __shared__ float cv[256 * TOPK];
    __shared__ int   ci[256 * TOPK];
    __shared__ float rv[256];
    __shared__ int   ri[256];
    __shared__ float selv[TOPK];
    __shared__ int   seli[TOPK];

    const int t = threadIdx.x;

    // per-thread sorted top-16 (registers; all indices constant after unroll)
    float lv[TOPK];
    int   li[TOPK];
    #pragma unroll
    for (int j = 0; j < TOPK; ++j) { lv[j] = __builtin_huge_valf(); li[j] = 0; }

    for (int i = t; i < N; i += 256) {
        float v = d2[i];
        if (v < lv[TOPK - 1]) {
            lv[TOPK - 1] = v; li[TOPK - 1] = i;
            #pragma unroll
            for (int j = TOPK - 1; j > 0; --j) {
                if (lv[j] < lv[j - 1]) {
                    float tv = lv[j]; lv[j] = lv[j - 1]; lv[j - 1] = tv;
                    int   ti = li[j]; li[j] = li[j - 1]; li[j - 1] = ti;
                }
            }
        }
    }

    #pragma unroll
    for (int j = 0; j < TOPK; ++j) { cv[t * TOPK + j] = lv[j]; ci[t * TOPK + j] = li[j]; }
    __syncthreads();

    // 16 rounds of block-wide argmin over the 4096 candidates
    for (int s = 0; s < TOPK; ++s) {
        float bv = __builtin_huge_valf();
        int   bp = 0;
        #pragma unroll
        for (int j = 0; j < TOPK; ++j) {
            int p = j * 256 + t;                // strided coverage of all 4096 slots
            float v = cv[p];
            if (v < bv) { bv = v; bp = p; }
        }
        rv[t] = bv; ri[t] = bp;
        __syncthreads();
        for (int off = 128; off > 0; off >>= 1) {
            if (t < off) {
                if (rv[t + off] < rv[t]) { rv[t] = rv[t + off]; ri[t] = ri[t + off]; }
            }
            __syncthreads();
        }
        if (t == 0) {
            int p = ri[0];
            selv[s] = rv[0];
            seli[s] = ci[p];
            cv[p] = __builtin_huge_valf();      // mask winner
        }
        __syncthreads();
    }

    if (t < TOPK) {
        // fold back the rank-invariant ||s||^2 term and take sqrt
        float s2 = 0.0f;
        for (int k = 0; k < D; ++k) { float x = samples[k]; s2 = fmaf(x, x, s2); }
        float dd = selv[t] + s2;
        if (dd < 0.0f) dd = 0.0f;
        ((int*)out)[t]  = seli[t];
        out[TOPK + t]   = sqrtf(dd);
    }
}

extern "C" void kernel_launch(void* const* d_in, const int* in_sizes, int n_in,
                              void* d_out, int out_size, void* d_ws, size_t ws_size,
                              hipStream_t stream) {
    const float* samples = (const float*)d_in[0];   // [1, 512] f32
    const float* mapv    = (const float*)d_in[1];   // [N, 512] f32
    (void)d_in; (void)n_in; (void)ws_size; (void)out_size;

    const int N      = in_sizes[1] / D;             // 100000 (multiple of 16)
    const int ntiles = N / 16;                      // 6250
    float* ws = (float*)d_ws;                       // N floats of dist^2-proxy

    dim3 blk(256);                                  // 8 waves -> 8 tiles per block
    dim3 grd((ntiles + 7) / 8);
    som_dist2_kernel<<<grd, blk, 0, stream>>>(samples, mapv, ws, ntiles);
    som_topk_kernel<<<1, 256, 0, stream>>>(samples, ws, N, (float*)d_out);
}